// BCE_sigmoid_58119497450044
// MI455X (gfx1250) — compile-verified
//
#include <hip/hip_runtime.h>
#include <hip/hip_bf16.h>

#define NCH 23
constexpr int NBLK = 2048;   // block partials in d_ws (8 KB)

// Reference weights: w0 = 1/(W+1), w1 = 1 - w0 = W/(W+1).
// Kernel needs 1/w0 = W+1 and 1/w1 = (W+1)/W, baked at compile time.
#define I0(w) ((float)(1.0 + (w)))
#define I1(w) ((float)((1.0 + (w)) / (w)))

#define W_LIST(F) \
  F(0.0012597430655963838), F(0.0004919313290455535), F(0.0021106513104319356), \
  F(0.0007678117365508301), F(0.004719881670572202),  F(0.000372272357115554),  \
  F(0.029090425620315438),  F(0.010056339432617042),  F(0.0034817436971298467), \
  F(0.0003057951504877765), F(0.003995280118329428),  F(8.808229878180519e-05), \
  F(0.012070598793438699),  F(0.016788818533845208),  F(0.0017832510677901316), \
  F(0.0008758371973209686), F(0.0005933090691529143), F(0.0031992155689617922), \
  F(0.003212511010287348),  F(0.0016685778863572154), F(0.0009356666832859684), \
  F(0.0010985358395240233), F(0.00103372056306194)

__constant__ float C_INV0[NCH] = { W_LIST(I0) };
__constant__ float C_INV1[NCH] = { W_LIST(I1) };

// ---------------------------------------------------------------------------
// Kernel 1: streaming weighted-BCE partial reduction.
// Grid-stride over float4/int4 chunks (global_load_b128), one v_log_f32 per
// element, per-channel weight via LDS table, block tree-reduce to d_ws.
// ---------------------------------------------------------------------------
__global__ __launch_bounds__(256) void bce_partial_kernel(
    const float* __restrict__ x, const int* __restrict__ lab,
    float* __restrict__ partial, int n4, int n) {
  __shared__ float w0s[NCH];
  __shared__ float w1s[NCH];
  __shared__ float red[256];

  const int tid = threadIdx.x;
  if (tid < NCH) {
    w0s[tid] = C_INV0[tid];
    w1s[tid] = C_INV1[tid];
  }
  __syncthreads();

  const int gtid   = blockIdx.x * blockDim.x + tid;
  const int stride = gridDim.x * blockDim.x;

  float acc = 0.0f;

  const float4* __restrict__ x4 = (const float4*)x;
  const int4*   __restrict__ l4 = (const int4*)lab;

  for (int i = gtid; i < n4; i += stride) {
    const float4 xv = x4[i];     // global_load_b128
    const int4   lv = l4[i];     // global_load_b128
    int c = (i * 4) % NCH;       // magic-multiply, once per chunk

    const float xx[4] = { xv.x, xv.y, xv.z, xv.w };
    const int   ll[4] = { lv.x, lv.y, lv.z, lv.w };
#pragma unroll
    for (int j = 0; j < 4; ++j) {
      const bool  t  = ll[j] > 0;
      const float v  = t ? xx[j] : 1.0f - xx[j];   // only one log term survives
      const float* wp = t ? w1s : w0s;             // pointer-select -> one ds_load
      const float w  = wp[c];
      acc = __builtin_fmaf(w, __logf(v), acc);     // v_log_f32 + fma
      ++c; if (c >= NCH) c = 0;
    }
  }

  // scalar tail (robustness; n is divisible by 4 for this problem)
  for (int e = n4 * 4 + gtid; e < n; e += stride) {
    const bool  t = lab[e] > 0;
    const float v = t ? x[e] : 1.0f - x[e];
    const float* wp = t ? w1s : w0s;
    acc = __builtin_fmaf(wp[e % NCH], __logf(v), acc);
  }

  // block tree reduction in LDS
  red[tid] = acc;
  __syncthreads();
  for (int s = 128; s > 0; s >>= 1) {
    if (tid < s) red[tid] += red[tid + s];
    __syncthreads();
  }
  if (tid == 0) partial[blockIdx.x] = red[0];
}

// ---------------------------------------------------------------------------
// Kernel 2: 256-thread final reduction.
// Each thread gathers 8 partials (unrolled -> 8 in-flight loads, one wait),
// LDS tree-reduces 256 -> 32, then EVERY wave redundantly reduces the 32
// values with one V_WMMA_F32_16X16X4_F32 (A = lane values in a 16x4 f32 tile,
// B = ones => D[m][n] = s[m] + s[m+16]). EXEC is all-ones in every wave around
// the WMMA, so the ISA restriction is honored without divergence; only wave-0
// lanes 0/16 publish the column-0 row sums, and thread 0 writes the output.
// ---------------------------------------------------------------------------
typedef float v2f __attribute__((ext_vector_type(2)));
typedef float v8f __attribute__((ext_vector_type(8)));

__global__ __launch_bounds__(256) void bce_final_kernel(
    const float* __restrict__ partial, float* __restrict__ out, float inv_total) {
  __shared__ float red[256];
  __shared__ float red2[2];
  const int tid = threadIdx.x;

  float s = 0.0f;
#pragma unroll
  for (int i = 0; i < NBLK / 256; ++i) s += partial[tid + i * 256];

  red[tid] = s;
  __syncthreads();
#pragma unroll
  for (int k = 128; k >= 32; k >>= 1) {
    if (tid < k) red[tid] += red[tid + k];
    __syncthreads();
  }

  // red[0..31] hold the 32 surviving partials; all waves compute identically.
  const float v = red[tid & 31];
  v2f a; a.x = v;    a.y = 0.0f;
  v2f b; b.x = 1.0f; b.y = 1.0f;
  v8f c = {};
  c = __builtin_amdgcn_wmma_f32_16x16x4_f32(
      /*neg_a=*/false, a, /*neg_b=*/false, b,
      /*c_mod=*/(short)0, c, /*reuse_a=*/false, /*reuse_b=*/false);
  const float t = c[0] + c[1] + c[2] + c[3] + c[4] + c[5] + c[6] + c[7];

  if (tid == 0)  red2[0] = t;   // rows 0..7  of column 0
  if (tid == 16) red2[1] = t;   // rows 8..15 of column 0
  __syncthreads();

  if (tid == 0) {
    // loss = sum(w * bce) / (n*c); bce = -log(selected prob), acc carried +log
    out[0] = -(red2[0] + red2[1]) * inv_total;
  }
}

// ---------------------------------------------------------------------------
extern "C" void kernel_launch(void* const* d_in, const int* in_sizes, int n_in,
                              void* d_out, int out_size, void* d_ws, size_t ws_size,
                              hipStream_t stream) {
  const float* x   = (const float*)d_in[0];
  const int*   lab = (const int*)d_in[1];
  float*       out = (float*)d_out;

  const int n  = in_sizes[0];      // B*C = 23,000,000
  const int n4 = n / 4;

  float* partials = (float*)d_ws;

  bce_partial_kernel<<<NBLK, 256, 0, stream>>>(x, lab, partials, n4, n);
  bce_final_kernel<<<1, 256, 0, stream>>>(partials, out, 1.0f / (float)n);
}